// Clustering_58428735094995
// MI455X (gfx1250) — compile-verified
//
#include <hip/hip_runtime.h>
#include <hip/hip_bf16.h>

typedef float v2f __attribute__((ext_vector_type(2)));
typedef float v8f __attribute__((ext_vector_type(8)));

#define HW    (512 * 1024)
#define NE    8
#define BATCH 16
#define TILE  1024
#define PS    1028   // padded LDS row stride (floats) to spread DS banks

// LDS byte-offset map (single block, red[] overlaps predS after final barrier)
#define OFF_PRED  0                            // float[8*PS]  = 32896 B
#define OFF_SQ    (OFF_PRED + NE * PS * 4)     // float[1024]  =  4096 B
#define OFF_LAB   (OFF_SQ + TILE * 4)          // int[1024]    =  4096 B
#define OFF_ZERO  (OFF_LAB + TILE * 4)         // float[132]   =   528 B (zeros)
#define OFF_ONE   (OFF_ZERO + 528)             // float[132]   =   528 B (ones)
#define SMEM_SZ   (OFF_ONE + 528)

// ---------------------------------------------------------------------------
// Pass 1: per-(batch, chunk) segmented reduction via V_WMMA_F32_16X16X4_F32.
// Wave-persistent 16x16 f32 accumulator D:
//   rows 0..7 : sums[e][label]   (A rows = pred[e] of 4 pixels)
//   row  8    : sumsq[label]     (A row  = sum_e pred^2 per pixel)
//   row  9    : counts[label]    (A row  = 1.0, fed from a 1.0-filled region)
//   cols 1..4 : labels 1..4 (one-hot B); label 0 handled analytically (mu=0)
// Inner loop is branchless AND add-free: every lane advances its LDS address
// by a uniform 16 B/iter (constant rows walk replicated 1.0/0.0 regions), so
// the unroller emits pure immediate-offset ds_load_b64 off one base register.
// ---------------------------------------------------------------------------
__global__ __launch_bounds__(256)
void cluster_partial(const float* __restrict__ pred,
                     const int*   __restrict__ lab,
                     float*       __restrict__ part,
                     int tilesPerChunk)
{
    __shared__ __align__(16) char smem[SMEM_SZ];
    float* predS = (float*)(smem + OFF_PRED);
    float* sqS   = (float*)(smem + OFF_SQ);
    int*   labS  = (int*)  (smem + OFF_LAB);
    float* red   = (float*)(smem + OFF_PRED);   // reused after last barrier

    const int tid  = threadIdx.x;
    const int wave = tid >> 5;
    const int lane = tid & 31;
    const int b      = blockIdx.y;
    const int chunk  = blockIdx.x;
    const int nchunk = gridDim.x;

    const float* predB = pred + (size_t)b * NE * HW;
    const int*   labB  = lab  + (size_t)b * HW;
    const size_t chunkBase = (size_t)chunk * (size_t)tilesPerChunk * TILE;

    // Fill the constant regions once (covered by the first tile barrier).
    for (int i = tid; i < 132; i += 256) {
        ((float*)(smem + OFF_ZERO))[i] = 0.0f;
        ((float*)(smem + OFF_ONE))[i]  = 1.0f;
    }

    // --- Per-lane addressing, computed once; uniform 16 B/iter stride ---
    const int m = lane & 15;        // A row / B col within half-wave
    const int g = lane >> 4;        // K-pair: 0 -> K0/K1, 1 -> K2/K3
    const int pByte = (wave * 128 + g * 2) * 4;   // byte offset of first pixel

    int aBase;
    if (m < 8)       aBase = OFF_PRED + m * PS * 4 + pByte;  // pred row e=m
    else if (m == 8) aBase = OFF_SQ + pByte;                 // sum_e pred^2
    else if (m == 9) aBase = OFF_ONE + g * 8;                // 1.0 -> counts
    else             aBase = OFF_ZERO + g * 8;               // 0.0 rows
    const int labBase = OFF_LAB + pByte;
    const int cmpVal  = (m == 0) ? 0x40000000 : m;           // drop label 0

    v8f c = {};   // persistent WMMA accumulator

    for (int t = 0; t < tilesPerChunk; ++t) {
        const size_t base = chunkBase + (size_t)t * TILE;
        __syncthreads();

        // --- Stage tile: 256 threads x 4 pixels, coalesced b128 ---
        const int p4 = tid * 4;
        float sq0 = 0.f, sq1 = 0.f, sq2 = 0.f, sq3 = 0.f;
#pragma unroll
        for (int e = 0; e < NE; ++e) {
            float4 v = *(const float4*)(predB + (size_t)e * HW + base + p4);
            *(float4*)&predS[e * PS + p4] = v;
            sq0 += v.x * v.x; sq1 += v.y * v.y;
            sq2 += v.z * v.z; sq3 += v.w * v.w;
        }
        *(float4*)&sqS[p4] = make_float4(sq0, sq1, sq2, sq3);
        *(int4*)&labS[p4]  = *(const int4*)(labB + base + p4);
        __syncthreads();

        // --- 128 pixels per wave: 32 chained f32 WMMAs, immediate offsets ---
#pragma unroll 8
        for (int it = 0; it < 32; ++it) {
            const v2f  a  = *(const v2f*) (smem + aBase   + it * 16);
            const int2 lp = *(const int2*)(smem + labBase + it * 16);
            v2f bm;
            bm.x = (lp.x == cmpVal) ? 1.0f : 0.0f;
            bm.y = (lp.y == cmpVal) ? 1.0f : 0.0f;
            c = __builtin_amdgcn_wmma_f32_16x16x4_f32(
                    false, a, false, bm, (short)0, c, false, false);
        }
    }

    // --- Cross-wave reduction of the 8 accumulators (fixed order) ---
    __syncthreads();
#pragma unroll
    for (int j = 0; j < 8; ++j) red[tid * 8 + j] = c[j];
    __syncthreads();

    if (tid < 40) {
        const int l = tid / 10 + 1;   // label 1..4
        const int j = tid % 10;       // 0..7 sums[e], 8 sumsq, 9 count
        float s = 0.f;
#pragma unroll
        for (int w = 0; w < 8; ++w) {
            const int idx = (j < 8) ? ((w * 32 + l) * 8 + j)              // rows 0..7
                                    : ((w * 32 + 16 + l) * 8 + (j - 8));  // rows 8,9
            s += red[idx];
        }
        part[(size_t)(b * nchunk + chunk) * 40 + tid] = s;
    }
}

// ---------------------------------------------------------------------------
// Pass 2: reduce chunk partials, compute L_var + L_dist epilogue, one scalar.
// ---------------------------------------------------------------------------
__global__ __launch_bounds__(256)
void cluster_final(const float* __restrict__ part,
                   float*       __restrict__ out,
                   int nchunk)
{
    __shared__ float tot[BATCH * 40];
    __shared__ float lvS[BATCH];
    __shared__ float ldS[BATCH];

    const int tid = threadIdx.x;
    for (int i = tid; i < BATCH * 40; i += 256) {
        const int b = i / 40, item = i % 40;
        float s = 0.f;
        for (int ch = 0; ch < nchunk; ++ch)
            s += part[(size_t)(b * nchunk + ch) * 40 + item];
        tot[i] = s;
    }
    __syncthreads();

    if (tid < BATCH) {
        const float* tb = &tot[tid * 40];
        float mu[5][NE];
#pragma unroll
        for (int e = 0; e < NE; ++e) mu[0][e] = 0.f;   // label 0: mu == 0

        float Lv = 0.f;
        for (int l = 1; l < 5; ++l) {
            const float* r = tb + (l - 1) * 10;
            const float cnt = r[9];
            const float inv = cnt > 0.f ? 1.f / cnt : 0.f;
            float msq = 0.f;
#pragma unroll
            for (int e = 0; e < NE; ++e) {
                const float mm = r[e] * inv;
                mu[l][e] = mm;
                msq += mm * mm;
            }
            const float frob = r[8] - cnt * msq;
            if (frob > 0.f) {
                const float n = sqrtf(frob);
                if (n > 0.5f) { const float d = n - 0.5f; Lv += d * d; }
            }
        }
        Lv *= 0.25f;   // / C, C = 4

        float Ld = 0.f;
        for (int i = 0; i < 4; ++i) {
            for (int j = 0; j < 4; ++j) {
                if (i == j) continue;
                float dsq = 0.f;
#pragma unroll
                for (int e = 0; e < NE; ++e) {
                    const float d = mu[i][e] - mu[j][e];
                    dsq += d * d;
                }
                if (dsq > 0.f) {
                    const float d = sqrtf(dsq);
                    const float h = 3.0f - d;
                    if (h > 0.f) Ld += h * h;
                } else {
                    Ld += 9.0f;   // DELTA_D^2 when dsq == 0 off-diagonal
                }
            }
        }
        lvS[tid] = Lv;
        ldS[tid] = Ld;
    }
    __syncthreads();

    if (tid == 0) {
        float a = 0.f, d = 0.f;
        for (int i = 0; i < BATCH; ++i) { a += lvS[i]; d += ldS[i]; }
        out[0] = (a + d) * (1.0f / BATCH);
    }
}

// ---------------------------------------------------------------------------
extern "C" void kernel_launch(void* const* d_in, const int* in_sizes, int n_in,
                              void* d_out, int out_size, void* d_ws, size_t ws_size,
                              hipStream_t stream)
{
    const float* pred = (const float*)d_in[0];
    // d_in[1] (binary_label) is redundant: implied by instance_label > 0.
    const int*   lab  = (const int*)d_in[2];
    float* out  = (float*)d_out;
    float* part = (float*)d_ws;

    // Pick chunk count from available workspace (needs 16*nc*40 floats).
    int nc = 64;
    while (nc > 1 && (size_t)(BATCH * nc * 40 * sizeof(float)) > ws_size) nc >>= 1;
    const int tilesPerChunk = HW / (nc * TILE);

    dim3 grid(nc, BATCH, 1);
    cluster_partial<<<grid, 256, 0, stream>>>(pred, lab, part, tilesPerChunk);
    cluster_final<<<1, 256, 0, stream>>>(part, out, nc);
}